// GENERICLayer_44341242363969
// MI455X (gfx1250) — compile-verified
//
#include <hip/hip_runtime.h>
#include <hip/hip_bf16.h>

// CDNA5 (gfx1250) fused kernel for the GENERIC phase-space layer.
//
// Math (simplified from the reference):
//   h      = z @ W1^T + b1                         (8192x128)
//   g      = W2_row * silu'(h)                     (elementwise)
//   grad_S = g @ W1                                (8192x64)
//   a      = z @ Wf^T                              (8192x64)
//   out    = [z[:,32:], -z[:,:32]] + a*a*grad_S    (eigh of diag matrix is identity-like)
//
// ~335 MFLOP over ~4.2 MB of traffic -> latency bound. Weights (49KB) live in
// LDS, each wave owns 16 rows, all GEMMs run on V_WMMA_F32_16X16X4_F32 (exact
// f32, matches reference precision).

typedef float v2f __attribute__((ext_vector_type(2)));
typedef float v8f __attribute__((ext_vector_type(8)));

#define NROWS 8192
#define DD 64      // state dim
#define HH 128     // hidden dim
#define WAVES_PER_BLOCK 4

__global__ __launch_bounds__(WAVES_PER_BLOCK * 32)
void generic_phase_kernel(const float* __restrict__ z,
                          const float* __restrict__ W1,
                          const float* __restrict__ b1,
                          const float* __restrict__ W2,
                          const float* __restrict__ Wf,
                          float* __restrict__ out)
{
    // ---- LDS: weights shared by all waves + per-wave g-staging slab ----
    __shared__ float sW1[HH * DD];                 // 32 KB, row-major [h][d]
    __shared__ float sWf[DD * DD];                 // 16 KB, row-major [dout][din]
    __shared__ float sB1[HH];
    __shared__ float sW2[HH];
    __shared__ float sG[WAVES_PER_BLOCK][16 * HH]; // 32 KB, per-wave 16x128 row-major

    const int tid  = threadIdx.x;        // 0..127
    const int wave = tid >> 5;           // 0..3
    const int lane = tid & 31;           // wave32
    const int lg   = lane >> 4;          // lane group: 0 => K{0,1}, 1 => K{2,3}
    const int ln   = lane & 15;          // row (A/C) or column (B/C) selector

    const int rowBase = (blockIdx.x * WAVES_PER_BLOCK + wave) * 16;

    // Prefetch this wave's z tile while we fill LDS (global_prefetch_b8).
    __builtin_prefetch(z + (size_t)rowBase * DD, 0, 0);

    // Cooperative weight loads (coalesced, once per block, reused by 64 rows).
    for (int i = tid; i < HH * DD; i += WAVES_PER_BLOCK * 32) sW1[i] = W1[i];
    for (int i = tid; i < DD * DD; i += WAVES_PER_BLOCK * 32) sWf[i] = Wf[i];
    if (tid < HH) { sB1[tid] = b1[tid]; sW2[tid] = W2[tid]; }
    __syncthreads();

    // ---- Preload A-fragments of z (16x64 tile, 16 K-steps of 4) ----------
    // f32 A 16x4 layout: lanes 0-15 rows M, v0/v1 = K0/K1; lanes 16-31 = K2/K3.
    v2f za[16];
    {
        const float* zrow = z + (size_t)(rowBase + ln) * DD;
        #pragma unroll
        for (int kk = 0; kk < 16; ++kk)
            za[kk] = *(const v2f*)(zrow + kk * 4 + lg * 2);
    }

    // ---- GEMM1: h = z @ W1^T  (B[k][n] = W1[n][k], contiguous float2) ----
    v8f h[8];
    #pragma unroll
    for (int nt = 0; nt < 8; ++nt) {
        const int n = nt * 16 + ln;
        v8f c = {};
        #pragma unroll
        for (int kk = 0; kk < 16; ++kk) {
            const int k = kk * 4 + lg * 2;
            v2f bfrag = *(const v2f*)(&sW1[n * DD + k]);
            c = __builtin_amdgcn_wmma_f32_16x16x4_f32(
                    false, za[kk], false, bfrag, (short)0, c, false, false);
        }
        h[nt] = c;
    }

    // ---- g = W2 * silu'(h + b1); scatter C-layout -> row-major LDS slab ---
    // C/D layout: VGPR v, lanes 0-15 -> M=v, lanes 16-31 -> M=v+8; N = nt*16+ln.
    float* gs = &sG[wave][0];
    #pragma unroll
    for (int nt = 0; nt < 8; ++nt) {
        const int n   = nt * 16 + ln;
        const float w2   = sW2[n];
        const float bias = sB1[n];
        #pragma unroll
        for (int v = 0; v < 8; ++v) {
            const int m = v + lg * 8;
            const float x = h[nt][v] + bias;
            const float s = 1.0f / (1.0f + __expf(-x));            // sigmoid
            const float g = w2 * (s * (1.0f + x * (1.0f - s)));    // W2*silu'
            gs[m * HH + n] = g;
        }
    }
    // Wave-private slab; DS ops from one wave are in-order -> no barrier.

    // ---- GEMM2: grad_S = g @ W1  (A from LDS slab; B[k][n] = W1[k][n]) ----
    v8f gradS[4];
    #pragma unroll
    for (int nt = 0; nt < 4; ++nt) {
        const int n = nt * 16 + ln;
        v8f c = {};
        #pragma unroll
        for (int kk = 0; kk < 32; ++kk) {
            const int k = kk * 4 + lg * 2;
            v2f afrag = *(const v2f*)(&gs[ln * HH + k]);
            v2f bfrag;
            bfrag.x = sW1[k * DD + n];
            bfrag.y = sW1[(k + 1) * DD + n];
            c = __builtin_amdgcn_wmma_f32_16x16x4_f32(
                    false, afrag, false, bfrag, (short)0, c, false, false);
        }
        gradS[nt] = c;
    }

    // ---- GEMM3: a = z @ Wf^T  (reuse za fragments; B[k][n] = Wf[n][k]) ----
    v8f amat[4];
    #pragma unroll
    for (int nt = 0; nt < 4; ++nt) {
        const int n = nt * 16 + ln;
        v8f c = {};
        #pragma unroll
        for (int kk = 0; kk < 16; ++kk) {
            const int k = kk * 4 + lg * 2;
            v2f bfrag = *(const v2f*)(&sWf[n * DD + k]);
            c = __builtin_amdgcn_wmma_f32_16x16x4_f32(
                    false, za[kk], false, bfrag, (short)0, c, false, false);
        }
        amat[nt] = c;
    }

    // ---- Epilogue: out = symplectic(z) + a*a*grad_S -----------------------
    #pragma unroll
    for (int nt = 0; nt < 4; ++nt) {
        const int d    = nt * 16 + ln;
        const int dr   = (d < 32) ? d + 32 : d - 32;
        const float sg = (d < 32) ? 1.0f : -1.0f;
        #pragma unroll
        for (int v = 0; v < 8; ++v) {
            const int m = v + lg * 8;
            const size_t row = (size_t)(rowBase + m);
            const float rev = sg * z[row * DD + dr];
            const float av  = amat[nt][v];
            out[row * DD + d] = rev + av * av * gradS[nt][v];
        }
    }
}

extern "C" void kernel_launch(void* const* d_in, const int* in_sizes, int n_in,
                              void* d_out, int out_size, void* d_ws, size_t ws_size,
                              hipStream_t stream) {
    (void)in_sizes; (void)n_in; (void)out_size; (void)d_ws; (void)ws_size;
    const float* z  = (const float*)d_in[0];
    const float* W1 = (const float*)d_in[1];
    const float* b1 = (const float*)d_in[2];
    const float* W2 = (const float*)d_in[3];
    // d_in[4] = b2: gradient w.r.t. z does not depend on it.
    const float* Wf = (const float*)d_in[5];
    float* out = (float*)d_out;

    const int rows_per_block = WAVES_PER_BLOCK * 16;      // 64
    dim3 grid(NROWS / rows_per_block);                    // 128 blocks
    dim3 block(WAVES_PER_BLOCK * 32);                     // 128 threads (4 waves)
    generic_phase_kernel<<<grid, block, 0, stream>>>(z, W1, b1, W2, Wf, out);
}